// SupConLoss_32435593019556
// MI455X (gfx1250) — compile-verified
//
#include <hip/hip_runtime.h>
#include <math.h>

// Problem constants (match reference)
#define B_SZ    4096
#define D_SZ    256
#define TWO_B   8192
#define THREE_B 12288
#define TEMP_INV (1.0f / 0.07f)
#define EPS_V    1e-6f
#define COS_EPS  1e-8f

#define LDS_STRIDE 260   // 256 + 4 floats pad -> conflict-free frag loads, keeps 16B align

typedef __attribute__((ext_vector_type(2))) float v2f;
typedef __attribute__((ext_vector_type(4))) float v4f;
typedef __attribute__((ext_vector_type(8))) float v8f;

// ---------------------------------------------------------------------------
// Async staging: `rows` x 256-float panel -> LDS (stride 260) via
// GLOBAL_LOAD_ASYNC_TO_LDS_B128 (ASYNCcnt-tracked, no VGPR data movement).
// 16B per lane per instruction; 256 threads -> rows/4 iterations.
// NOT unrolled: keeps one live address set (avoids VGPR-MSB pressure).
// ---------------------------------------------------------------------------
__device__ __forceinline__ void stage_panel_async(const float* __restrict__ srcBase,
                                                  int rows, float* ldsPanel, int t) {
    const int iters = rows >> 2;   // rows*256 floats / (256 thr * 4 floats)
#pragma unroll 1
    for (int q = 0; q < iters; ++q) {
        int fi  = (t + 256 * q) * 4;   // flat float index into rows x 256 panel
        int row = fi >> 8;
        int col = fi & 255;
        unsigned int       ldsaddr = (unsigned int)(size_t)(ldsPanel + row * LDS_STRIDE + col);
        unsigned long long gaddr   = (unsigned long long)(size_t)(srcBase + fi);
        asm volatile("global_load_async_to_lds_b128 %0, %1, off"
                     :: "v"(ldsaddr), "v"(gaddr) : "memory");
    }
}

// Panels are 64/128-row aligned and B_SZ is a multiple of both, so a panel
// never straddles the z1|z2 boundary -> one base pointer per panel.
__device__ __forceinline__ const float* panel_base(const float* z1, const float* z2,
                                                   int baseRow) {
    return (baseRow < B_SZ) ? z1 + (size_t)baseRow * D_SZ
                            : z2 + (size_t)(baseRow - B_SZ) * D_SZ;
}

__device__ __forceinline__ void wait_async_le16() {
    asm volatile("s_wait_asynccnt 0x10" ::: "memory");
}
__device__ __forceinline__ void wait_async_0() {
    asm volatile("s_wait_asynccnt 0x0" ::: "memory");
}

// ---------------------------------------------------------------------------
// Kernel 1: per-row L2 norms of f = [z1; z2; zn]  (one wave32 per row)
// ---------------------------------------------------------------------------
__global__ __launch_bounds__(256) void norms_kernel(const float* __restrict__ z1,
                                                    const float* __restrict__ z2,
                                                    const float* __restrict__ zn,
                                                    float* __restrict__ norms) {
    int row  = (blockIdx.x * blockDim.x + threadIdx.x) >> 5;
    int lane = threadIdx.x & 31;
    if (row >= THREE_B) return;
    const float* src = (row < B_SZ)   ? z1 + (size_t)row * D_SZ
                     : (row < 2*B_SZ) ? z2 + (size_t)(row - B_SZ) * D_SZ
                                      : zn + (size_t)(row - 2*B_SZ) * D_SZ;
    v4f a = *(const v4f*)(src + lane * 8);
    v4f b = *(const v4f*)(src + lane * 8 + 4);
    float s = a.x*a.x + a.y*a.y + a.z*a.z + a.w*a.w
            + b.x*b.x + b.y*b.y + b.z*b.z + b.w*b.w;
#pragma unroll
    for (int m = 16; m >= 1; m >>= 1) s += __shfl_xor(s, m, 32);
    if (lane == 0) norms[row] = sqrtf(s);
}

// ---------------------------------------------------------------------------
// Kernel 2: positive logits P (concat order: d_pos[2B], d_neg[B], d_far[B])
// ---------------------------------------------------------------------------
__global__ __launch_bounds__(256) void pos_kernel(const float* __restrict__ z1,
                                                  const float* __restrict__ z2,
                                                  const float* __restrict__ zn,
                                                  const float* __restrict__ norms,
                                                  float* __restrict__ P) {
    int i    = (blockIdx.x * blockDim.x + threadIdx.x) >> 5;
    int lane = threadIdx.x & 31;
    if (i >= B_SZ) return;
    const float* a = z1 + (size_t)i * D_SZ;
    const float* b = z2 + (size_t)i * D_SZ;
    const float* c = zn + (size_t)i * D_SZ;
    float dab = 0.f, dbc = 0.f, dac = 0.f;
#pragma unroll
    for (int q = 0; q < 2; ++q) {
        int k = lane * 8 + q * 4;
        v4f va = *(const v4f*)(a + k);
        v4f vb = *(const v4f*)(b + k);
        v4f vc = *(const v4f*)(c + k);
        dab += va.x*vb.x + va.y*vb.y + va.z*vb.z + va.w*vb.w;
        dbc += vb.x*vc.x + vb.y*vc.y + vb.z*vc.z + vb.w*vc.w;
        dac += va.x*vc.x + va.y*vc.y + va.z*vc.z + va.w*vc.w;
    }
#pragma unroll
    for (int m = 16; m >= 1; m >>= 1) {
        dab += __shfl_xor(dab, m, 32);
        dbc += __shfl_xor(dbc, m, 32);
        dac += __shfl_xor(dac, m, 32);
    }
    if (lane == 0) {
        float n1 = norms[i], n2 = norms[B_SZ + i], nn = norms[2 * B_SZ + i];
        float pab = __expf((dab / fmaxf(n1 * n2, COS_EPS)) * TEMP_INV);
        float pbn = __expf((dbc / fmaxf(n2 * nn, COS_EPS)) * TEMP_INV);
        float pan = __expf((dac / fmaxf(n1 * nn, COS_EPS)) * TEMP_INV);
        P[i]            = pab;  // d_pos[i]   = sim(z1_i, z2_i)
        P[B_SZ + i]     = pbn;  // d_pos[B+i] = sim(z2_i, zn_i)
        P[2 * B_SZ + i] = pab;  // d_neg[i]   = sim(z2_i, z1_i) (symmetric)
        P[3 * B_SZ + i] = pan;  // d_far[i]   = sim(z1_i, zn_i)
    }
}

// ---------------------------------------------------------------------------
// Kernel 3: masked row sums of exp(cos_sim/T) over z@z^T (8192x8192), fused.
// Grid (64, 8): block owns 128 rows x 1024 cols; 16 inner tiles of 64 cols,
// double-buffered async DMA overlapped with WMMA compute.
// 8 waves in 4x2 layout; each wave: 32x32 via 2x2 accs of V_WMMA_F32_16X16X4.
// ---------------------------------------------------------------------------
__global__ __launch_bounds__(256) void simsum_kernel(const float* __restrict__ z1,
                                                     const float* __restrict__ z2,
                                                     const float* __restrict__ norms,
                                                     float* __restrict__ rowSum) {
    extern __shared__ float lds[];
    float* ldsA  = lds;                                     // 128 * 260
    float* ldsB0 = lds + 128 * LDS_STRIDE;                  //  64 * 260
    float* ldsB1 = ldsB0 + 64 * LDS_STRIDE;                 //  64 * 260
    float* ldsR  = ldsB1 + 64 * LDS_STRIDE;                 // 128

    const int t    = threadIdx.x;
    const int lane = t & 31;
    const int w    = t >> 5;        // wave 0..7
    const int wr   = w >> 1;        // 0..3 : 32-row group
    const int wc   = w & 1;         // 0..1 : 32-col group
    const int RB   = blockIdx.x * 128;
    const int CB   = blockIdx.y * 1024;

    if (t < 128) ldsR[t] = 0.f;

    stage_panel_async(panel_base(z1, z2, RB), 128, ldsA, t);   // row panel
    stage_panel_async(panel_base(z1, z2, CB), 64, ldsB0, t);   // col tile 0

    const int la = lane & 15;         // N / M index within 16
    const int kh = (lane >> 4) * 2;   // K sub-offset 0 or 2
    const int halfsel = lane >> 4;    // C-fragment M-half

#pragma unroll 1
    for (int ct = 0; ct < 16; ++ct) {
        const int CTB = CB + ct * 64;
        float* ldsB = (ct & 1) ? ldsB1 : ldsB0;

        // Prefetch next col tile into the other buffer (overlaps with compute),
        // then drain everything older than that batch and sync all waves.
        if (ct < 15) {
            stage_panel_async(panel_base(z1, z2, CTB + 64), 64,
                              (ct & 1) ? ldsB0 : ldsB1, t);
            wait_async_le16();
        } else {
            wait_async_0();
        }
        __syncthreads();   // current tile fully in LDS (all waves' batches done)

        v8f acc[2][2] = {};
        const float* pa0 = ldsA + (wr * 32 + la) * LDS_STRIDE + kh;
        const float* pb0 = ldsB + (wc * 32 + la) * LDS_STRIDE + kh;

#pragma unroll 4
        for (int k = 0; k < D_SZ; k += 4) {
            v2f af[2], bf[2];
#pragma unroll
            for (int mt = 0; mt < 2; ++mt)
                af[mt] = *(const v2f*)(pa0 + mt * 16 * LDS_STRIDE + k);
#pragma unroll
            for (int nt = 0; nt < 2; ++nt)
                bf[nt] = *(const v2f*)(pb0 + nt * 16 * LDS_STRIDE + k);
#pragma unroll
            for (int mt = 0; mt < 2; ++mt)
#pragma unroll
                for (int nt = 0; nt < 2; ++nt)
                    acc[mt][nt] = __builtin_amdgcn_wmma_f32_16x16x4_f32(
                        false, af[mt], false, bf[nt],
                        (short)0, acc[mt][nt], false, false);
        }

        // Epilogue: cos-sim (fast rcp), exp, diagonal mask, row reduction
        float ncv[2];
#pragma unroll
        for (int nt = 0; nt < 2; ++nt)
            ncv[nt] = norms[CTB + wc * 32 + nt * 16 + la];

#pragma unroll
        for (int mt = 0; mt < 2; ++mt) {
#pragma unroll
            for (int v = 0; v < 8; ++v) {
                int r  = RB + wr * 32 + mt * 16 + v + 8 * halfsel;
                float nr = norms[r];
                float e = 0.f;
#pragma unroll
                for (int nt = 0; nt < 2; ++nt) {
                    int c = CTB + wc * 32 + nt * 16 + la;
                    float s   = __builtin_amdgcn_rcpf(fmaxf(nr * ncv[nt], COS_EPS));
                    float ev  = __expf(acc[mt][nt][v] * s * TEMP_INV);
                    bool masked = (c == r) || (c == r + B_SZ) || (c + B_SZ == r);
                    e += masked ? 0.f : ev;
                }
#pragma unroll
                for (int m = 8; m >= 1; m >>= 1) e += __shfl_xor(e, m, 32);
                if (la == 0)
                    atomicAdd(&ldsR[wr * 32 + mt * 16 + v + 8 * halfsel], e);
            }
        }
        __syncthreads();   // all waves done with this tile's buffer
    }

    if (t < 128) atomicAdd(&rowSum[RB + t], ldsR[t]);
}

// ---------------------------------------------------------------------------
// Kernel 4: loss = (1/B) * sum_k [ log(rowSum[k]+eps) - 0.5*(log P[2k]+log P[2k+1]) ]
// ---------------------------------------------------------------------------
__global__ __launch_bounds__(256) void finalize_kernel(const float* __restrict__ rowSum,
                                                       const float* __restrict__ P,
                                                       float* __restrict__ out) {
    int k    = blockIdx.x * blockDim.x + threadIdx.x;
    int lane = threadIdx.x & 31;
    float term = 0.f;
    if (k < TWO_B) {
        term = __logf(rowSum[k] + EPS_V)
             - 0.5f * (__logf(P[2 * k]) + __logf(P[2 * k + 1]));
    }
#pragma unroll
    for (int m = 16; m >= 1; m >>= 1) term += __shfl_xor(term, m, 32);
    if (lane == 0) atomicAdd(out, term * (1.0f / (float)B_SZ));
}

// ---------------------------------------------------------------------------
extern "C" void kernel_launch(void* const* d_in, const int* in_sizes, int n_in,
                              void* d_out, int out_size, void* d_ws, size_t ws_size,
                              hipStream_t stream) {
    const float* z1 = (const float*)d_in[0];
    const float* z2 = (const float*)d_in[1];
    const float* zn = (const float*)d_in[2];
    float* out = (float*)d_out;

    // workspace layout: norms[3B] | rowSum[2B] | P[4B]
    float* norms  = (float*)d_ws;
    float* rowSum = norms + THREE_B;
    float* P      = rowSum + TWO_B;

    hipMemsetAsync(out, 0, sizeof(float), stream);
    hipMemsetAsync(rowSum, 0, TWO_B * sizeof(float), stream);

    norms_kernel<<<THREE_B / 8, 256, 0, stream>>>(z1, z2, zn, norms);
    pos_kernel<<<B_SZ / 8, 256, 0, stream>>>(z1, z2, zn, norms, P);

    size_t ldsBytes = (size_t)(128 * LDS_STRIDE + 2 * 64 * LDS_STRIDE + 128) * sizeof(float);
    dim3 grid(TWO_B / 128, TWO_B / 1024);
    simsum_kernel<<<grid, 256, ldsBytes, stream>>>(z1, z2, norms, rowSum);

    finalize_kernel<<<(TWO_B + 255) / 256, 256, 0, stream>>>(rowSum, P, out);
}